// HungarianMatcher_61641370632782
// MI455X (gfx1250) — compile-verified
//
#include <hip/hip_runtime.h>

typedef __attribute__((ext_vector_type(2))) float v2f;
typedef __attribute__((ext_vector_type(8))) float v8f;

#define K_COST_CLASS 2.0f
#define K_COST_BBOX  5.0f
#define K_COST_GIOU  2.0f
#define K_ALPHA      0.25f
#define K_EPS        1e-8f

// ---------------------------------------------------------------------------
// Kernel 1: per-(query, class) focal cost table.
// tab[n*C + c] = pos - neg for logit[n, c].  Only 14400*42 = 604800 evals,
// amortizing all transcendental work instead of paying it per output element.
// ---------------------------------------------------------------------------
__global__ void matcher_class_tab_kernel(const float* __restrict__ logits,
                                         float* __restrict__ tab, int total) {
    int i = blockIdx.x * blockDim.x + threadIdx.x;
    if (i >= total) return;
    float x = logits[i];
    float p = 1.0f / (1.0f + __expf(-x));
    float omp = 1.0f - p;
    float pos = K_ALPHA * omp * omp * (-__logf(p + K_EPS));          // GAMMA = 2
    float neg = (1.0f - K_ALPHA) * p * p * (-__logf(omp + K_EPS));
    tab[i] = pos - neg;
}

// ---------------------------------------------------------------------------
// Kernel 2: one wave per 16x16 output tile. v_wmma_f32_16x16x4_f32 computes
// the cdist cross term (a . b) for the whole tile; VALU does GIoU + combine.
// Block = 4 waves sharing one row-tile (row data staged once in LDS).
// ---------------------------------------------------------------------------
__launch_bounds__(128)
__global__ void matcher_cost_tile_kernel(const float* __restrict__ boxes, // [N,4] cxcywh
                                         const float* __restrict__ tgt,   // [T,4] cxcywh
                                         const int*   __restrict__ tids,  // [T]
                                         const float* __restrict__ ctab,  // [N,C]
                                         float*       __restrict__ out,   // [N,T]
                                         int N, int T, int C) {
    // row-derived data: x0 y0 x1 y1 area sumsq (pad to 8 floats)
    __shared__ float srow[16][8];

    const int row0 = blockIdx.y * 16;
    const int lane = threadIdx.x & 31;
    const int wave = threadIdx.x >> 5;
    const int col0 = (blockIdx.x * 4 + wave) * 16;

    if (threadIdx.x < 16) {
        const float4 bb = ((const float4*)boxes)[row0 + threadIdx.x];
        srow[threadIdx.x][0] = bb.x - 0.5f * bb.z;
        srow[threadIdx.x][1] = bb.y - 0.5f * bb.w;
        srow[threadIdx.x][2] = bb.x + 0.5f * bb.z;
        srow[threadIdx.x][3] = bb.y + 0.5f * bb.w;
        srow[threadIdx.x][4] = bb.z * bb.w; // (x1-x0)*(y1-y0) == w*h
        srow[threadIdx.x][5] = bb.x * bb.x + bb.y * bb.y + bb.z * bb.z + bb.w * bb.w;
    }
    __syncthreads();

    const int hi = lane >> 4;   // 0: K=0,1 / rows 0-7 of D ; 1: K=2,3 / rows 8-15
    const int lo = lane & 15;
    const int k0 = hi * 2;

    // A fragment: 16x4 f32, lane L holds A[L&15][k0], A[L&15][k0+1]
    const int ma = row0 + lo;
    v2f afrag;
    afrag.x = boxes[ma * 4 + k0];
    afrag.y = boxes[ma * 4 + k0 + 1];

    // B fragment: 4x16 f32, lane L holds B[k0][col], B[k0+1][col]
    const int n = col0 + lo;
    v2f bfrag;
    bfrag.x = tgt[n * 4 + k0];
    bfrag.y = tgt[n * 4 + k0 + 1];

    v8f acc = {};
    // D[m][n] = sum_k boxes[m][k] * tgt[n][k]   (16x16 dot-products, K=4)
    acc = __builtin_amdgcn_wmma_f32_16x16x4_f32(
        /*neg_a=*/false, afrag, /*neg_b=*/false, bfrag,
        /*c_mod=*/(short)0, acc, /*reuse_a=*/false, /*reuse_b=*/false);

    // column-derived data (one column per lane, kept in registers)
    const float4 tb = ((const float4*)tgt)[n];
    const float bx0 = tb.x - 0.5f * tb.z, by0 = tb.y - 0.5f * tb.w;
    const float bx1 = tb.x + 0.5f * tb.z, by1 = tb.y + 0.5f * tb.w;
    const float areaB = tb.z * tb.w;
    const float ssB = tb.x * tb.x + tb.y * tb.y + tb.z * tb.z + tb.w * tb.w;
    const int tid_n = tids[n];

#pragma unroll
    for (int r = 0; r < 8; ++r) {
        const int mloc = r + hi * 8;      // D-matrix row layout (ISA 7.12.2)
        const int m = row0 + mloc;

        const float ax0 = srow[mloc][0], ay0 = srow[mloc][1];
        const float ax1 = srow[mloc][2], ay1 = srow[mloc][3];
        const float areaA = srow[mloc][4], ssA = srow[mloc][5];

        // cdist: sqrt(max(|a|^2 + |b|^2 - 2 a.b, 0))
        const float d2 = ssA + ssB - 2.0f * acc[r];
        const float cost_bbox = __builtin_amdgcn_sqrtf(fmaxf(d2, 0.0f));

        // GIoU
        const float ix0 = fmaxf(ax0, bx0), iy0 = fmaxf(ay0, by0);
        const float ix1 = fminf(ax1, bx1), iy1 = fminf(ay1, by1);
        const float iw = fmaxf(ix1 - ix0, 0.0f), ih = fmaxf(iy1 - iy0, 0.0f);
        const float inter = iw * ih;
        const float uni = areaA + areaB - inter;
        const float iou = inter * __builtin_amdgcn_rcpf(uni);
        const float ex0 = fminf(ax0, bx0), ey0 = fminf(ay0, by0);
        const float ex1 = fmaxf(ax1, bx1), ey1 = fmaxf(ay1, by1);
        const float ew = fmaxf(ex1 - ex0, 0.0f), eh = fmaxf(ey1 - ey0, 0.0f);
        const float ae = ew * eh;
        const float giou = iou - (ae - uni) * __builtin_amdgcn_rcpf(ae);

        const float cost_class = ctab[m * C + tid_n];

        out[(long)m * T + n] =
            K_COST_BBOX * cost_bbox + K_COST_CLASS * cost_class - K_COST_GIOU * giou;
    }
}

extern "C" void kernel_launch(void* const* d_in, const int* in_sizes, int n_in,
                              void* d_out, int out_size, void* d_ws, size_t ws_size,
                              hipStream_t stream) {
    const float* logits = (const float*)d_in[0]; // [bs, nq, C]
    const float* boxes  = (const float*)d_in[1]; // [bs, nq, 4]
    const float* tgt    = (const float*)d_in[2]; // [T, 4]
    const int*   tids   = (const int*)d_in[3];   // [T]
    float* out = (float*)d_out;

    const int N = in_sizes[1] / 4;       // bs*nq = 14400
    const int T = in_sizes[3];           // 3200
    const int C = in_sizes[0] / N;       // 42

    float* ctab = (float*)d_ws;          // N*C floats = ~2.4 MB

    const int total = N * C;
    matcher_class_tab_kernel<<<(total + 255) / 256, 256, 0, stream>>>(logits, ctab, total);

    dim3 grid(T / 64, N / 16);           // 4 waves/block, one 16x16 tile per wave
    matcher_cost_tile_kernel<<<grid, 128, 0, stream>>>(boxes, tgt, tids, ctab, out, N, T, C);
}